// GraphConvolution_60911226192170
// MI455X (gfx1250) — compile-verified
//
#include <hip/hip_runtime.h>

// GraphConvolution for MI455X (gfx1250, wave32):
//  k1: support = X @ W           -- f32 WMMA 16x16x4, LDS-staged tiles
//  k2: out[n][:] = bias          -- full re-init (harness poisons d_out)
//  k3: out[r][:] += v * support[c]  -- wave-per-edge SpMM, L2-resident f32 atomics

typedef float v2f __attribute__((ext_vector_type(2)));
typedef float v8f __attribute__((ext_vector_type(8)));

#define D   128   // D_IN == D_OUT
#define LDP 132   // padded LDS row stride (floats): keeps rows 16B-aligned and
                  // gives bank stride 4 dwords -> conflict-free ds_load_b64

__global__ void __launch_bounds__(256) gcn_gemm_wmma(
    const float* __restrict__ X, const float* __restrict__ W,
    float* __restrict__ support, int nRows)
{
    extern __shared__ float lds[];
    float* Xs = lds;               // [128][LDP]  local X tile
    float* Wt = lds + 128 * LDP;   // [128][LDP]  Wt[n][k] = W[k][n]

    const int t    = threadIdx.x;      // 0..255
    const int lane = t & 31;
    const int wave = t >> 5;           // 0..7
    const int row0 = blockIdx.x * 128; // first X row of this block

    // ---- stage 128x128 X tile: coalesced global_load_b128 -> padded LDS ----
    #pragma unroll
    for (int i = 0; i < 16; ++i) {
        int f4 = t + i * 256;          // 0..4095 float4 slots
        int r  = f4 >> 5;              // 32 float4 per row
        int c  = (f4 & 31) << 2;
        float4 v = make_float4(0.f, 0.f, 0.f, 0.f);
        int gr = row0 + r;
        if (gr < nRows) v = *(const float4*)(X + (size_t)gr * D + c);
        *(float4*)(Xs + r * LDP + c) = v;
    }
    // ---- stage W transposed: coalesced read, scattered LDS write (one-time) ----
    #pragma unroll
    for (int i = 0; i < 16; ++i) {
        int f4 = t + i * 256;
        int k  = f4 >> 5;
        int n  = (f4 & 31) << 2;
        float4 v = *(const float4*)(W + k * D + n);
        Wt[(n + 0) * LDP + k] = v.x;
        Wt[(n + 1) * LDP + k] = v.y;
        Wt[(n + 2) * LDP + k] = v.z;
        Wt[(n + 3) * LDP + k] = v.w;
    }
    __syncthreads();

    const int mTile = blockIdx.x * 8 + wave;   // 16-row output tile (wave-uniform)
    if (mTile * 16 >= nRows) return;           // uniform branch: EXEC stays all-1s

    const int half = lane >> 4;   // 0/1: which K-pair this lane holds
    const int m    = lane & 15;   // row (A) / col (B) within tile

    v8f acc[8] = {};              // 8 N-tiles of 16 -> full 16x128 strip
    const float* arow = Xs + (wave * 16 + m) * LDP;

    #pragma unroll 2
    for (int k0 = 0; k0 < D; k0 += 4) {
        // Issue ALL fragment loads for this k-step first, so the scheduler can
        // batch the ds_loads and overlap their latency with the WMMA chain
        // instead of a load->s_wait_dscnt 0->wmma lockstep.
        v2f a = *(const v2f*)(arow + k0 + 2 * half);       // a[v]=X[m][k0+2h+v]
        v2f b[8];
        #pragma unroll
        for (int j = 0; j < 8; ++j) {
            // b[j][v] = W[k0 + 2*half + v][j*16 + m]
            b[j] = *(const v2f*)(Wt + (j * 16 + m) * LDP + k0 + 2 * half);
        }
        #pragma unroll
        for (int j = 0; j < 8; ++j) {
            acc[j] = __builtin_amdgcn_wmma_f32_16x16x4_f32(
                /*neg_a=*/false, a, /*neg_b=*/false, b[j],
                /*c_mod=*/(short)0, acc[j],
                /*reuse_a=*/false, /*reuse_b=*/false);
        }
    }

    // C/D layout: VGPR v -> rows v (lanes 0-15) and v+8 (lanes 16-31), N = m
    float* outp = support + (size_t)mTile * 16 * D;
    #pragma unroll
    for (int j = 0; j < 8; ++j) {
        #pragma unroll
        for (int v = 0; v < 8; ++v) {
            outp[(v + 8 * half) * D + j * 16 + m] = acc[j][v];
        }
    }
}

// out[n][d] = bias[d]  (full init of d_out, float4 granularity)
__global__ void __launch_bounds__(256) gcn_bias_init(
    float* __restrict__ out, const float* __restrict__ bias, int nNodes)
{
    const float4* b4 = (const float4*)bias;
    float4* o4 = (float4*)out;
    int total = nNodes * (D / 4);
    for (int idx = blockIdx.x * blockDim.x + threadIdx.x; idx < total;
         idx += gridDim.x * blockDim.x) {
        o4[idx] = b4[idx & (D / 4 - 1)];
    }
}

// One wave per edge: lanes load 32 edges' (r,c,v) coalesced, broadcast via
// __shfl (wave32), gather support[c] as float4, atomic-add into out[r] (L2).
__global__ void __launch_bounds__(256) gcn_spmm_atomic(
    const float* __restrict__ support, const float* __restrict__ vals,
    const int* __restrict__ rows, const int* __restrict__ cols,
    float* __restrict__ out, int nEdges)
{
    const int lane = threadIdx.x & 31;
    const int wid  = (blockIdx.x * blockDim.x + threadIdx.x) >> 5;
    const int nW   = (gridDim.x * blockDim.x) >> 5;

    for (int base = wid * 32; base < nEdges; base += nW * 32) {
        int   e = base + lane;
        int   r = 0, c = 0;
        float v = 0.f;
        if (e < nEdges) { r = rows[e]; c = cols[e]; v = vals[e]; }
        int cnt = nEdges - base; if (cnt > 32) cnt = 32;

        for (int j = 0; j < cnt; ++j) {
            int   rj = __shfl(r, j, 32);
            int   cj = __shfl(c, j, 32);
            float vj = __shfl(v, j, 32);
            float4 s = *(const float4*)(support + (size_t)cj * D + lane * 4);
            float* dst = out + (size_t)rj * D + lane * 4;
            atomicAdd(dst + 0, vj * s.x);
            atomicAdd(dst + 1, vj * s.y);
            atomicAdd(dst + 2, vj * s.z);
            atomicAdd(dst + 3, vj * s.w);
        }
    }
}

extern "C" void kernel_launch(void* const* d_in, const int* in_sizes, int n_in,
                              void* d_out, int out_size, void* d_ws, size_t ws_size,
                              hipStream_t stream)
{
    const float* x     = (const float*)d_in[0];  // [N, 128]
    const float* w     = (const float*)d_in[1];  // [128, 128]
    const float* bias  = (const float*)d_in[2];  // [128]
    const float* evals = (const float*)d_in[3];  // [E]
    const int*   erows = (const int*)d_in[4];    // [E]
    const int*   ecols = (const int*)d_in[5];    // [E]
    float* out     = (float*)d_out;
    float* support = (float*)d_ws;               // [N, 128] scratch

    const int nNodes = in_sizes[0] / D;
    const int nEdges = in_sizes[3];

    const int mTiles     = (nNodes + 15) / 16;
    const int gemmBlocks = (mTiles + 7) / 8;     // 8 waves/block, 1 tile each
    const size_t ldsBytes = 2u * 128u * LDP * sizeof(float); // 132 KB

    gcn_gemm_wmma<<<gemmBlocks, 256, ldsBytes, stream>>>(x, w, support, nNodes);
    gcn_bias_init<<<2048, 256, 0, stream>>>(out, bias, nNodes);
    gcn_spmm_atomic<<<2048, 256, 0, stream>>>(support, evals, erows, ecols, out, nEdges);
}